// Behavior_67259187855641
// MI455X (gfx1250) — compile-verified
//
#include <hip/hip_runtime.h>
#include <math.h>

#define FD 5           // feature dim into both layers (F_IN == F_HID == 5)
#define NNODES 100000
#define NEDGES 6400000

typedef __attribute__((ext_vector_type(2))) float v2f;
typedef __attribute__((ext_vector_type(8))) float v8f;

__device__ __forceinline__ int imin(int a, int b) { return a < b ? a : b; }
__device__ __forceinline__ int imax(int a, int b) { return a > b ? a : b; }

// Guaranteed-native fp32 scatter-add (no CAS-loop fallback): VGLOBAL
// GLOBAL_ATOMIC_ADD_F32, no-return form, tracked by STOREcnt and drained by
// the implicit S_WAIT_IDLE at S_ENDPGM.
__device__ __forceinline__ void gatomic_add_f32(float* p, float v) {
    asm volatile("global_atomic_add_f32 %0, %1, off"
                 :
                 : "v"(p), "v"(v)
                 : "memory");
}

// ---------------------------------------------------------------- zero fill
__global__ void zero_f32(float* __restrict__ p, int n) {
    int i = blockIdx.x * blockDim.x + threadIdx.x;
    int stride = gridDim.x * blockDim.x;
    for (; i < n; i += stride) p[i] = 0.0f;
}

// ------------------------------------------------- edge gather/scatter pass
// agg[dst] += h[src] (FD floats), optionally deg[dst] += 1.
// Feature table (2MB) and agg/deg (2.4MB) are L2-resident (192MB L2), so the
// random gathers + fp32 scatter-atomics resolve in L2; HBM traffic is the
// streamed src/dst lists (~25.6MB each per pass).
template <bool ADD_DEG>
__global__ void edge_agg(const float* __restrict__ h,
                         const int* __restrict__ src,
                         const int* __restrict__ dst,
                         float* __restrict__ agg,
                         float* __restrict__ deg,
                         int n_edges) {
    int e = blockIdx.x * blockDim.x + threadIdx.x;
    if (e >= n_edges) return;
    int s = src[e];
    int d = dst[e];
    const float* hs = h + (long)s * FD;
    float* ad = agg + (long)d * FD;
#pragma unroll
    for (int k = 0; k < FD; ++k) {
        gatomic_add_f32(ad + k, hs[k]);
    }
    if (ADD_DEG) {
        gatomic_add_f32(deg + d, 1.0f);
    }
}

// ---------------------------------------------- node transform via WMMA f32
// out[m, n] = sigmoid( sum_k A[m,k] * Wcat[k,n] + b[n] )
//   A[m, 0..4]  = h[m, :]            (self path)
//   A[m, 5..9]  = agg[m, :] / deg[m] (mean-neighbor path)
//   Wcat        = [W_self ; W_neigh], K padded 10 -> 12, N padded F_OUT -> 16
// One 16-node tile per wave; 3 chained V_WMMA_F32_16X16X4_F32 (K = 3 x 4).
// Fragment element selection is clamped-load + select; any residual
// predication the compiler introduces re-converges before each WMMA, and the
// wave-uniform early exit keeps EXEC all-1s whenever a WMMA executes.
//
// ISA fragment layouts (CDNA5 ISA):
//   A 16x4 f32 : lane = 16*half + m, VGPR j holds A[m][2*half + j]
//   B 4x16 f32 : lane = 16*half + n, VGPR j holds B[2*half + j][n]
//   C 16x16    : lane = 16*half + n, VGPR r holds C[r + 8*half][n]
template <int F_OUT>
__global__ void node_wmma(const float* __restrict__ h,
                          const float* __restrict__ agg,
                          const float* __restrict__ deg,
                          const float* __restrict__ Wself,
                          const float* __restrict__ Wneigh,
                          const float* __restrict__ bias,
                          float* __restrict__ out,
                          int n_nodes) {
    const int lane  = threadIdx.x & 31;
    const int wave  = threadIdx.x >> 5;
    const int tile  = blockIdx.x * (blockDim.x >> 5) + wave;
    const int m0    = tile * 16;
    if (m0 >= n_nodes) return;          // wave-uniform exit: EXEC all-1s below

    const int row   = lane & 15;        // M for A, N for B/C
    const int khalf = lane >> 4;        // K half-select

    const int m  = m0 + row;
    const int mc = imin(m, n_nodes - 1);            // clamp loads, keep EXEC full
    const float invdeg = 1.0f / fmaxf(deg[mc], 1.0f);

    const int n  = row;
    const int nc = imin(n, F_OUT - 1);
    const bool nvalid = (n < F_OUT);

    v8f c = {};
#pragma unroll
    for (int chunk = 0; chunk < 3; ++chunk) {
        v2f a, b;
#pragma unroll
        for (int j = 0; j < 2; ++j) {
            const int k  = chunk * 4 + khalf * 2 + j;     // 0..11
            const int kh = imin(k, FD - 1);               // clamped self index
            const int kn = imin(imax(k - FD, 0), FD - 1); // clamped neigh index

            const float hv = h[(long)mc * FD + kh];
            const float nv = agg[(long)mc * FD + kn] * invdeg;
            const float av = (k < FD) ? hv : ((k < 2 * FD) ? nv : 0.0f);

            const float ws = Wself[kh * F_OUT + nc];
            const float wn = Wneigh[kn * F_OUT + nc];
            float bv = (k < FD) ? ws : ((k < 2 * FD) ? wn : 0.0f);
            bv = nvalid ? bv : 0.0f;

            a[j] = av;
            b[j] = bv;
        }
        // (neg_a, A, neg_b, B, c_mod, C, reuse_a, reuse_b)
        c = __builtin_amdgcn_wmma_f32_16x16x4_f32(false, a, false, b,
                                                  (short)0, c, false, false);
    }

    const float bn = bias[nc];
#pragma unroll
    for (int r = 0; r < 8; ++r) {
        const int mm = m0 + r + khalf * 8;
        if (mm < n_nodes && nvalid) {
            const float v = c[r] + bn;
            out[(long)mm * F_OUT + n] = 1.0f / (1.0f + __expf(-v));
        }
    }
}

// ------------------------------------------------------------------ driver
extern "C" void kernel_launch(void* const* d_in, const int* in_sizes, int n_in,
                              void* d_out, int out_size, void* d_ws, size_t ws_size,
                              hipStream_t stream) {
    (void)in_sizes; (void)n_in; (void)out_size; (void)ws_size;

    const float* x      = (const float*)d_in[0];
    const int*   src    = (const int*)d_in[1];
    const int*   dst    = (const int*)d_in[2];
    const float* Wself1 = (const float*)d_in[3];
    const float* Wneigh1= (const float*)d_in[4];
    const float* b1     = (const float*)d_in[5];
    const float* Wself2 = (const float*)d_in[6];
    const float* Wneigh2= (const float*)d_in[7];
    const float* b2     = (const float*)d_in[8];
    float* out = (float*)d_out;

    float* h   = (float*)d_ws;             // NNODES * FD
    float* agg = h + (size_t)NNODES * FD;  // NNODES * FD
    float* deg = agg + (size_t)NNODES * FD;// NNODES

    const int ZB = 256;
    const int EB = 256;
    const int edge_grid = (NEDGES + EB - 1) / EB;

    const int NODE_B = 256;                         // 8 waves -> 128 nodes/block
    const int tiles = (NNODES + 15) / 16;
    const int node_grid = (tiles + (NODE_B / 32) - 1) / (NODE_B / 32);

    // ---- layer 1 ----
    {
        const int nz = NNODES * FD + NNODES;        // agg + deg contiguous
        zero_f32<<<(nz + ZB - 1) / ZB, ZB, 0, stream>>>(agg, nz);
    }
    edge_agg<true><<<edge_grid, EB, 0, stream>>>(x, src, dst, agg, deg, NEDGES);
    node_wmma<FD><<<node_grid, NODE_B, 0, stream>>>(x, agg, deg, Wself1, Wneigh1,
                                                    b1, h, NNODES);

    // ---- layer 2 (graph unchanged; reuse deg) ----
    {
        const int nz = NNODES * FD;
        zero_f32<<<(nz + ZB - 1) / ZB, ZB, 0, stream>>>(agg, nz);
    }
    edge_agg<false><<<edge_grid, EB, 0, stream>>>(h, src, dst, agg, deg, NEDGES);
    node_wmma<10><<<node_grid, NODE_B, 0, stream>>>(h, agg, deg, Wself2, Wneigh2,
                                                    b2, out, NNODES);
}